// TtMixtralAttention_23880018166323
// MI455X (gfx1250) — compile-verified
//
#include <hip/hip_runtime.h>

typedef __attribute__((ext_vector_type(2))) float v2f;
typedef __attribute__((ext_vector_type(4))) float v4f;
typedef __attribute__((ext_vector_type(8))) float v8f;

static constexpr int Hdim  = 4096;   // hidden
static constexpr int NHn   = 32;     // query heads
static constexpr int NKVn  = 8;      // kv heads
static constexpr int Gn    = 4;      // NH / NKV
static constexpr int Dn    = 128;    // head dim
static constexpr int SWn   = 4096;   // cache window
static constexpr int Bn    = 32;     // batch
static constexpr int NQKV  = Hdim + 2 * NKVn * Dn;  // 6144

// ---------------------------------------------------------------------------
// Generic skinny GEMM  C(32 x N) = A(32 x K) * B(K x N) * scale, fp32 WMMA.
// One wave per 16x16 output tile. grid = (N/16, 2), block = 32.
// ---------------------------------------------------------------------------
__global__ __launch_bounds__(32)
void gemm_m32_wmma(const float* __restrict__ A, int lda,
                   const float* __restrict__ B, int ldb,
                   float* __restrict__ C, int ldc, int K, float scale)
{
    const int lane = threadIdx.x;
    const int l    = lane & 15;
    const int hw   = lane >> 4;                 // 0/1: K-half
    const long n0  = (long)blockIdx.x * 16;
    const int  m0  = blockIdx.y * 16;

    const float* Ap = A + (long)(m0 + l) * lda;
    const float* Bp = B + n0 + l;

    v8f c = {};
    for (int k0 = 0; k0 < K; k0 += 4) {
        const int ka = k0 + 2 * hw;
        v2f a = *(const v2f*)(Ap + ka);                       // A[m][ka], A[m][ka+1]
        v2f b = { Bp[(long)ka * ldb], Bp[(long)(ka + 1) * ldb] };
        c = __builtin_amdgcn_wmma_f32_16x16x4_f32(
                false, a, false, b, (short)0, c, false, false);
    }
    float* Cp = C + (long)(m0 + hw * 8) * ldc + n0 + l;
#pragma unroll
    for (int r = 0; r < 8; ++r)
        Cp[(long)r * ldc] = c[r] * scale;
}

// ---------------------------------------------------------------------------
// Batched rotation: heads 0..31 -> q_rot (scaled by D^-0.5), heads 32..39 -> k_rot.
// Per head: (32 x 128) @ rot(128 x 128). grid = (16 tiles, 40 heads), block = 32.
// ---------------------------------------------------------------------------
__global__ __launch_bounds__(32)
void rot_wmma(const float* __restrict__ qkv, const float* __restrict__ rot,
              float* __restrict__ qrot, float* __restrict__ knew)
{
    const int head = blockIdx.y;
    const int tile = blockIdx.x;
    const int n0   = (tile & 7) * 16;
    const int m0   = (tile >> 3) * 16;
    const int lane = threadIdx.x;
    const int l    = lane & 15;
    const int hw   = lane >> 4;

    const float* A; float* C; int ldc; float scale;
    if (head < NHn) {                                   // query head
        A = qkv + head * Dn;
        C = qrot + head * Dn;
        ldc = NHn * Dn;
        scale = 0.088388347648318447f;                  // 1/sqrt(128)
    } else {                                            // kv head
        const int h = head - NHn;
        A = qkv + Hdim + h * Dn;
        C = knew + (long)h * Bn * Dn;                   // knew[h][b][d]
        ldc = Dn;
        scale = 1.0f;
    }

    const float* Ap = A + (long)(m0 + l) * NQKV;
    const float* Bp = rot + n0 + l;

    v8f c = {};
    for (int k0 = 0; k0 < Dn; k0 += 4) {
        const int ka = k0 + 2 * hw;
        v2f a = *(const v2f*)(Ap + ka);
        v2f b = { Bp[ka * Dn], Bp[(ka + 1) * Dn] };
        c = __builtin_amdgcn_wmma_f32_16x16x4_f32(
                false, a, false, b, (short)0, c, false, false);
    }
    float* Cp = C + (long)(m0 + hw * 8) * ldc + n0 + l;
#pragma unroll
    for (int r = 0; r < 8; ++r)
        Cp[(long)r * ldc] = c[r] * scale;
}

// ---------------------------------------------------------------------------
// Attention, flash-style split over the sequence.
// grid = (NKV, B, 4 splits), block = 256 (8 waves).
// Streams cache_k/cache_v (the 1 GB) with non-temporal loads; row p==current_pos
// is taken from the fresh k/v (inputs are never mutated).
// Emits per-split (m, s, o_unnormalized) records for the combine pass.
// ---------------------------------------------------------------------------
__global__ __launch_bounds__(256)
void attn_partial(const float* __restrict__ qrot, const float* __restrict__ knew,
                  const float* __restrict__ qkv,
                  const float* __restrict__ cache_k, const float* __restrict__ cache_v,
                  const int* __restrict__ sp, const int* __restrict__ cp,
                  float* __restrict__ pbuf)
{
    const int h = blockIdx.x, b = blockIdx.y, split = blockIdx.z;
    const int t = threadIdx.x;
    const int L   = min(sp[0] + 1, SWn);      // layer_slice
    const int cur = cp[0];
    const int Lc  = (L + 3) >> 2;             // keys per split (<= 1024)
    const int p0  = split * Lc;
    const int p1  = min(L, p0 + Lc);
    const int n   = p1 - p0;

    __shared__ float sq[Gn * Dn];             // 2 KB  q tile
    __shared__ float sc[Gn * 1024];           // 16 KB scores for this split
    __shared__ float part[8 * Gn * Dn];       // 16 KB AV partials (per wave-group)
    __shared__ float red[Gn * 64];            // 1 KB  reductions
    __shared__ float stat[2 * Gn];            // m, sum per g

    for (int i = t; i < Gn * Dn; i += 256)
        sq[i] = qrot[(long)b * (NHn * Dn) + h * (Gn * Dn) + i];
    __syncthreads();

    // ---- scores: s[g][p] = q[g] . k[p]  (q already scaled) ----
    for (int pp = t; pp < n; pp += 256) {
        const int p = p0 + pp;
        const float* krow = (p == cur)
            ? (knew + ((long)h * Bn + b) * Dn)
            : (cache_k + (((long)h * Bn + b) * SWn + p) * Dn);
        float a0 = 0.f, a1 = 0.f, a2 = 0.f, a3 = 0.f;
#pragma unroll 8
        for (int d = 0; d < Dn; d += 4) {
            v4f k4 = __builtin_nontemporal_load((const v4f*)(krow + d));
            a0 += k4.x*sq[0*Dn+d] + k4.y*sq[0*Dn+d+1] + k4.z*sq[0*Dn+d+2] + k4.w*sq[0*Dn+d+3];
            a1 += k4.x*sq[1*Dn+d] + k4.y*sq[1*Dn+d+1] + k4.z*sq[1*Dn+d+2] + k4.w*sq[1*Dn+d+3];
            a2 += k4.x*sq[2*Dn+d] + k4.y*sq[2*Dn+d+1] + k4.z*sq[2*Dn+d+2] + k4.w*sq[2*Dn+d+3];
            a3 += k4.x*sq[3*Dn+d] + k4.y*sq[3*Dn+d+1] + k4.z*sq[3*Dn+d+2] + k4.w*sq[3*Dn+d+3];
        }
        sc[0*1024+pp] = a0; sc[1*1024+pp] = a1;
        sc[2*1024+pp] = a2; sc[3*1024+pp] = a3;
    }
    __syncthreads();

    // ---- local softmax stats per g (64 threads per g) ----
    const int g = t >> 6, sub = t & 63;
    float lm = -__builtin_inff();
    for (int pp = sub; pp < n; pp += 64) lm = fmaxf(lm, sc[g*1024 + pp]);
    red[g*64 + sub] = lm;
    __syncthreads();
    for (int s = 32; s > 0; s >>= 1) {
        if (sub < s) red[g*64+sub] = fmaxf(red[g*64+sub], red[g*64+sub+s]);
        __syncthreads();
    }
    const float mg = red[g*64];
    __syncthreads();
    float ls = 0.f;
    for (int pp = sub; pp < n; pp += 64) {
        const float e = __expf(sc[g*1024+pp] - mg);
        sc[g*1024+pp] = e;
        ls += e;
    }
    red[g*64 + sub] = ls;
    __syncthreads();
    for (int s = 32; s > 0; s >>= 1) {
        if (sub < s) red[g*64+sub] += red[g*64+sub+s];
        __syncthreads();
    }
    if (sub == 0) { stat[g] = mg; stat[Gn + g] = red[g*64]; }
    __syncthreads();

    // ---- o_unnorm[g][d] = sum_p e[g][p] * V[p][d] ----
    // wave w (pg) handles p = pg, pg+8, ...; lanes cover one full 512B V row.
    const int d4 = (t & 31) * 4;
    const int pg = t >> 5;
    float acc[Gn][4];
#pragma unroll
    for (int gg = 0; gg < Gn; ++gg)
#pragma unroll
        for (int c2 = 0; c2 < 4; ++c2) acc[gg][c2] = 0.f;

    for (int pp = pg; pp < n; pp += 8) {
        const int p = p0 + pp;
        const float* vrow = (p == cur)
            ? (qkv + (long)b * NQKV + Hdim + NKVn * Dn + h * Dn)
            : (cache_v + (((long)h * Bn + b) * SWn + p) * Dn);
        v4f v4 = __builtin_nontemporal_load((const v4f*)(vrow + d4));
#pragma unroll
        for (int gg = 0; gg < Gn; ++gg) {
            const float w = sc[gg*1024 + pp];           // LDS broadcast
            acc[gg][0] += w * v4.x; acc[gg][1] += w * v4.y;
            acc[gg][2] += w * v4.z; acc[gg][3] += w * v4.w;
        }
    }
#pragma unroll
    for (int gg = 0; gg < Gn; ++gg)
#pragma unroll
        for (int c2 = 0; c2 < 4; ++c2)
            part[(pg*Gn + gg)*Dn + d4 + c2] = acc[gg][c2];
    __syncthreads();

    // ---- write split record: [m(4), s(4), o(4*128)] = 520 floats ----
    float* rec = pbuf + (((long)h * Bn + b) * 4 + split) * 520;
    if (t < 128) {
        const int gg = t >> 5; const int dd = (t & 31) * 4;
        float s0=0.f, s1=0.f, s2=0.f, s3=0.f;
#pragma unroll
        for (int w8 = 0; w8 < 8; ++w8) {
            const float* pr = &part[(w8*Gn + gg)*Dn + dd];
            s0 += pr[0]; s1 += pr[1]; s2 += pr[2]; s3 += pr[3];
        }
        float* o = rec + 8 + gg * Dn + dd;
        o[0]=s0; o[1]=s1; o[2]=s2; o[3]=s3;
    }
    if (t < Gn) { rec[t] = stat[t]; rec[4 + t] = stat[Gn + t]; }
}

// ---------------------------------------------------------------------------
// Combine the 4 splits: M = max m_i, rescale sums/outputs, normalize, write
// aout[b][(h*G+g)*D + d]. grid = (NKV, B), block = 128.
// ---------------------------------------------------------------------------
__global__ __launch_bounds__(128)
void attn_combine(const float* __restrict__ pbuf, float* __restrict__ aout)
{
    const int h = blockIdx.x, b = blockIdx.y, t = threadIdx.x;
    const int g = t >> 5; const int d4 = (t & 31) * 4;
    const float* rec = pbuf + ((long)h * Bn + b) * 4 * 520;

    float M = -__builtin_inff();
#pragma unroll
    for (int i = 0; i < 4; ++i) M = fmaxf(M, rec[i*520 + g]);
    float S = 0.f, w[4];
#pragma unroll
    for (int i = 0; i < 4; ++i) {
        const float wi = __expf(rec[i*520 + g] - M);    // exp(-inf)=0 for empty splits
        w[i] = wi;
        S += wi * rec[i*520 + 4 + g];
    }
    const float inv = 1.0f / S;

    float o0=0.f, o1=0.f, o2=0.f, o3=0.f;
#pragma unroll
    for (int i = 0; i < 4; ++i) {
        const float* op = rec + i*520 + 8 + g * Dn + d4;
        o0 += w[i]*op[0]; o1 += w[i]*op[1]; o2 += w[i]*op[2]; o3 += w[i]*op[3];
    }
    float* out = aout + (long)b * (NHn * Dn) + (h * Gn + g) * Dn + d4;
    out[0]=o0*inv; out[1]=o1*inv; out[2]=o2*inv; out[3]=o3*inv;
}

// ---------------------------------------------------------------------------
extern "C" void kernel_launch(void* const* d_in, const int* in_sizes, int n_in,
                              void* d_out, int out_size, void* d_ws, size_t ws_size,
                              hipStream_t stream)
{
    const float* x   = (const float*)d_in[0];
    const float* wq  = (const float*)d_in[1];
    const float* wk  = (const float*)d_in[2];
    const float* wv  = (const float*)d_in[3];
    const float* wo  = (const float*)d_in[4];
    const float* rot = (const float*)d_in[5];
    const float* ck  = (const float*)d_in[6];
    const float* cv  = (const float*)d_in[7];
    const int*   sp  = (const int*)d_in[8];   // start_pos
    const int*   cp  = (const int*)d_in[9];   // current_pos

    float* ws   = (float*)d_ws;
    float* qkv  = ws;                          // 32*6144
    float* qrot = qkv  + Bn * NQKV;            // 32*4096
    float* knew = qrot + Bn * NHn * Dn;        // 8*32*128
    float* aout = knew + NKVn * Bn * Dn;       // 32*4096
    float* pbuf = aout + Bn * NHn * Dn;        // 256*4*520  (~4.1 MB total ws)

    // QKV projection (streams 100 MB of weights through WMMA)
    gemm_m32_wmma<<<dim3(256, 2), 32, 0, stream>>>(x, Hdim, wq, NHn*Dn,  qkv,           NQKV, Hdim, 1.0f);
    gemm_m32_wmma<<<dim3(64, 2),  32, 0, stream>>>(x, Hdim, wk, NKVn*Dn, qkv + Hdim,    NQKV, Hdim, 1.0f);
    gemm_m32_wmma<<<dim3(64, 2),  32, 0, stream>>>(x, Hdim, wv, NKVn*Dn, qkv + Hdim + NKVn*Dn, NQKV, Hdim, 1.0f);

    // rotation + q scaling (40 batched 32x128x128 WMMA GEMMs)
    rot_wmma<<<dim3(16, 40), 32, 0, stream>>>(qkv, rot, qrot, knew);

    // attention: 1024 workgroups stream the 1 GB KV cache
    attn_partial<<<dim3(NKVn, Bn, 4), 256, 0, stream>>>(qrot, knew, qkv, ck, cv, sp, cp, pbuf);
    attn_combine<<<dim3(NKVn, Bn), 128, 0, stream>>>(pbuf, aout);

    // output projection
    gemm_m32_wmma<<<dim3(256, 2), 32, 0, stream>>>(aout, NHn*Dn, wo, Hdim, (float*)d_out, Hdim, NHn*Dn, 1.0f);
}